// PointNet2ssg_29532195127603
// MI455X (gfx1250) — compile-verified
//
#include <hip/hip_runtime.h>
#include <hip/hip_bf16.h>

typedef float v2f __attribute__((ext_vector_type(2)));
typedef float v8f __attribute__((ext_vector_type(8)));

#define BN_EPS 1e-5f

// ---------------------------------------------------------------- transpose (B,3,N) -> (B,N,3)
__global__ void pn2_transpose_xyz(const float* __restrict__ in, float* __restrict__ out,
                                  int B, int N) {
  long t = (long)blockIdx.x * blockDim.x + threadIdx.x;
  long total = (long)B * N * 3;
  if (t >= total) return;
  int c = (int)(t % 3);
  long bn = t / 3;
  int n = (int)(bn % N);
  int b = (int)(bn / N);
  out[t] = in[((long)b * 3 + c) * N + n];
}

// ---------------------------------------------------------------- farthest point sampling
__global__ __launch_bounds__(256) void pn2_fps(const float* __restrict__ xyz,
                                               int* __restrict__ out, int N, int S) {
  __shared__ float dist[4096];
  __shared__ float rmax[256];
  __shared__ int rarg[256];
  __shared__ int far_s;
  int b = blockIdx.x, tid = threadIdx.x;
  const float* P = xyz + (long)b * N * 3;
  for (int n = tid; n < N; n += 256) dist[n] = 1e10f;
  if (tid == 0) far_s = 0;
  __syncthreads();
  for (int it = 0; it < S; ++it) {
    int far = far_s;
    if (tid == 0) out[b * S + it] = far;
    float cx = P[far * 3 + 0], cy = P[far * 3 + 1], cz = P[far * 3 + 2];
    float bmax = -1.0f;
    int barg = 0x7fffffff;
    for (int n = tid; n < N; n += 256) {
      float dx = P[n * 3 + 0] - cx, dy = P[n * 3 + 1] - cy, dz = P[n * 3 + 2] - cz;
      float d = dx * dx + dy * dy + dz * dz;
      float dn = fminf(dist[n], d);
      dist[n] = dn;
      if (dn > bmax) { bmax = dn; barg = n; }
    }
    rmax[tid] = bmax; rarg[tid] = barg;
    __syncthreads();
    for (int s = 128; s > 0; s >>= 1) {
      if (tid < s) {
        float m2 = rmax[tid + s]; int a2 = rarg[tid + s];
        if (m2 > rmax[tid] || (m2 == rmax[tid] && a2 < rarg[tid])) {
          rmax[tid] = m2; rarg[tid] = a2;
        }
      }
      __syncthreads();
    }
    if (tid == 0) far_s = rarg[0];
    __syncthreads();
  }
}

// ---------------------------------------------------------------- gather xyz by index
__global__ void pn2_gather3(const float* __restrict__ xyz, const int* __restrict__ idx,
                            float* __restrict__ out, int N, int S, int B) {
  long t = (long)blockIdx.x * blockDim.x + threadIdx.x;
  long total = (long)B * S * 3;
  if (t >= total) return;
  int c = (int)(t % 3);
  long bs = t / 3;
  int s = (int)(bs % S);
  int b = (int)(bs / S);
  out[t] = xyz[((long)b * N + idx[b * S + s]) * 3 + c];
}

// ---------------------------------------------------------------- ball query (first-K in index order)
__global__ void pn2_ball(const float* __restrict__ xyz, const float* __restrict__ nxyz,
                         int* __restrict__ idx, int B, int N, int S, int K, float r2) {
  long t = (long)blockIdx.x * blockDim.x + threadIdx.x;
  if (t >= (long)B * S) return;
  int b = (int)(t / S);
  const float* P = xyz + (long)b * N * 3;
  const float* q = nxyz + t * 3;
  int* o = idx + t * K;
  float qx = q[0], qy = q[1], qz = q[2];
  int cnt = 0, first = 0;
  for (int n = 0; n < N; ++n) {
    float dx = P[n * 3 + 0] - qx, dy = P[n * 3 + 1] - qy, dz = P[n * 3 + 2] - qz;
    if (dx * dx + dy * dy + dz * dz <= r2) {
      if (cnt == 0) first = n;
      o[cnt++] = n;
      if (cnt == K) break;
    }
  }
  for (; cnt < K; ++cnt) o[cnt] = first;
}

// ---------------------------------------------------------------- grouping with padded row stride
__global__ void pn2_group(const float* __restrict__ xyz, const float* __restrict__ nxyz,
                          const float* __restrict__ pfeat, const int* __restrict__ idx,
                          float* __restrict__ X, int B, int N, int S, int K, int Cp,
                          int Cpad) {
  int C = 3 + Cp;
  long t = (long)blockIdx.x * blockDim.x + threadIdx.x;
  long total = (long)B * S * K * Cpad;
  if (t >= total) return;
  int c = (int)(t % Cpad);
  long r = t / Cpad;               // (b*S+s)*K + k
  int k = (int)(r % K);
  long bs = r / K;
  int b = (int)(bs / S);
  float v = 0.0f;
  if (c < C) {
    int point = idx[bs * K + k];
    if (c < 3) v = xyz[((long)b * N + point) * 3 + c] - nxyz[bs * 3 + c];
    else       v = pfeat[((long)b * N + point) * Cp + (c - 3)];
  }
  X[t] = v;
}

// ---------------------------------------------------------------- zero-pad weights: Win[O,C] -> Wout[Opad,Cpad]
__global__ void pn2_pad_w(const float* __restrict__ Win, float* __restrict__ Wout,
                          int O, int C, int Opad, int Cpad) {
  long t = (long)blockIdx.x * blockDim.x + threadIdx.x;
  long total = (long)Opad * Cpad;
  if (t >= total) return;
  int k = (int)(t % Cpad);
  int o = (int)(t / Cpad);
  Wout[t] = (o < O && k < C) ? Win[(long)o * C + k] : 0.0f;
}

// ---------------------------------------------------------------- WMMA fp32 GEMM: Y[M,O] = X[M,C] * W[.,C]^T + bias
// Preconditions: M % 16 == 0, C % 4 == 0, ceil(O/16) % NB == 0,
//                W has >= ceil(O/16)*16 rows of stride C.
// One wave32 computes a 16 x (NB*16) tile: one A fragment feeds NB V_WMMA_F32_16X16X4_F32
// per K-step (NB-fold reuse of the streamed activation operand).
template <int NB>
__global__ __launch_bounds__(256) void pn2_gemm_bias(const float* __restrict__ X,
                                                     const float* __restrict__ W,
                                                     const float* __restrict__ bias,
                                                     float* __restrict__ Y,
                                                     long M, int C, int O) {
  int lane = threadIdx.x & 31;
  int wib  = threadIdx.x >> 5;
  int tilesNB = ((O + 15) >> 4) / NB;
  long numTiles = (M >> 4) * (long)tilesNB;
  long tile = (long)blockIdx.x * (blockDim.x >> 5) + wib;
  if (tile >= numTiles) return;          // wave-uniform: EXEC stays all-ones for WMMA
  int tnb = (int)(tile % tilesNB);
  long tm = tile / tilesNB;
  long row0 = tm << 4;
  int col0 = tnb * (NB * 16);
  int r = lane & 15, half = lane >> 4;
  const float* Ap = X + (row0 + r) * (long)C + half * 2;
  const float* Bp = W + (long)(col0 + r) * C + half * 2;
  v8f acc[NB];
#pragma unroll
  for (int j = 0; j < NB; ++j) acc[j] = v8f{};
#pragma unroll 2
  for (int c0 = 0; c0 < C; c0 += 4) {
    v2f a = *(const v2f*)(Ap + c0);
#pragma unroll
    for (int j = 0; j < NB; ++j) {
      v2f b = *(const v2f*)(Bp + (long)j * 16 * C + c0);
      acc[j] = __builtin_amdgcn_wmma_f32_16x16x4_f32(false, a, false, b, (short)0,
                                                     acc[j], false, false);
    }
  }
#pragma unroll
  for (int j = 0; j < NB; ++j) {
    int col = col0 + j * 16 + r;
    if (col < O) {
      float bv = bias ? bias[col] : 0.0f;
      float* Yp = Y + row0 * O + col;
#pragma unroll
      for (int g = 0; g < 8; ++g)        // C/D layout: VGPR g -> M=g (lanes<16) / M=g+8
        Yp[(long)(g + half * 8) * O] = acc[j][g] + bv;
    }
  }
}

// ---------------------------------------------------------------- per-channel mean/var (population)
__global__ __launch_bounds__(256) void pn2_bn_stats(const float* __restrict__ Y,
                                                    float* __restrict__ mean,
                                                    float* __restrict__ var,
                                                    long M, int O) {
  __shared__ float s1[256], s2[256];
  int o = blockIdx.x, tid = threadIdx.x;
  float a = 0.0f, b = 0.0f;
  for (long m = tid; m < M; m += 256) {
    float v = Y[m * O + o];
    a += v; b += v * v;
  }
  s1[tid] = a; s2[tid] = b;
  __syncthreads();
  for (int s = 128; s > 0; s >>= 1) {
    if (tid < s) { s1[tid] += s1[tid + s]; s2[tid] += s2[tid + s]; }
    __syncthreads();
  }
  if (tid == 0) {
    float mu = s1[0] / (float)M;
    mean[o] = mu;
    var[o] = s2[0] / (float)M - mu * mu;
  }
}

// ---------------------------------------------------------------- BN normalize + ReLU (in place)
__global__ void pn2_bn_relu(float* __restrict__ Y, const float* __restrict__ mean,
                            const float* __restrict__ var, const float* __restrict__ g,
                            const float* __restrict__ beta, long M, int O) {
  long t = (long)blockIdx.x * blockDim.x + threadIdx.x;
  if (t >= M * (long)O) return;
  int o = (int)(t % O);
  float v = (Y[t] - mean[o]) * rsqrtf(var[o] + BN_EPS) * g[o] + beta[o];
  Y[t] = v > 0.0f ? v : 0.0f;
}

// ---------------------------------------------------------------- max over nsample
__global__ void pn2_maxpool(const float* __restrict__ Y, float* __restrict__ out,
                            long BS, int K, int C) {
  long t = (long)blockIdx.x * blockDim.x + threadIdx.x;
  if (t >= BS * C) return;
  int c = (int)(t % C);
  long bs = t / C;
  const float* p = Y + bs * K * (long)C + c;
  float m = p[0];
  for (int k = 1; k < K; ++k) m = fmaxf(m, p[(long)k * C]);
  out[t] = m;
}

// ---------------------------------------------------------------- 3-NN + inverse-distance weights
__global__ void pn2_knn3(const float* __restrict__ xyz1, const float* __restrict__ xyz2,
                         int* __restrict__ idx, float* __restrict__ wgt,
                         int B, int N1, int N2) {
  long t = (long)blockIdx.x * blockDim.x + threadIdx.x;
  if (t >= (long)B * N1) return;
  int b = (int)(t / N1);
  const float* q = xyz1 + t * 3;
  const float* P = xyz2 + (long)b * N2 * 3;
  float qx = q[0], qy = q[1], qz = q[2];
  float d0 = 1e30f, d1 = 1e30f, d2 = 1e30f;
  int i0 = 0, i1 = 0, i2 = 0;
  for (int n = 0; n < N2; ++n) {
    float dx = P[n * 3 + 0] - qx, dy = P[n * 3 + 1] - qy, dz = P[n * 3 + 2] - qz;
    float d = dx * dx + dy * dy + dz * dz;
    if (d < d0)      { d2 = d1; i2 = i1; d1 = d0; i1 = i0; d0 = d; i0 = n; }
    else if (d < d1) { d2 = d1; i2 = i1; d1 = d; i1 = n; }
    else if (d < d2) { d2 = d; i2 = n; }
  }
  float w0 = 1.0f / (d0 + 1e-8f), w1 = 1.0f / (d1 + 1e-8f), w2 = 1.0f / (d2 + 1e-8f);
  float s = w0 + w1 + w2;
  idx[t * 3 + 0] = i0; idx[t * 3 + 1] = i1; idx[t * 3 + 2] = i2;
  wgt[t * 3 + 0] = w0 / s; wgt[t * 3 + 1] = w1 / s; wgt[t * 3 + 2] = w2 / s;
}

// ---------------------------------------------------------------- X = concat(p1, interp(p2))
__global__ void pn2_interp_concat(const float* __restrict__ p1, const float* __restrict__ p2,
                                  const int* __restrict__ idx, const float* __restrict__ wgt,
                                  float* __restrict__ X, int B, int N1, int N2,
                                  int C1, int C2) {
  int C = C1 + C2;
  long t = (long)blockIdx.x * blockDim.x + threadIdx.x;
  if (t >= (long)B * N1 * C) return;
  int c = (int)(t % C);
  long r = t / C;                        // b*N1 + n
  int b = (int)(r / N1);
  float v;
  if (c < C1) {
    v = p1[r * C1 + c];
  } else {
    int cc = c - C1;
    const float* P2 = p2 + (long)b * N2 * C2 + cc;
    v = wgt[r * 3 + 0] * P2[(long)idx[r * 3 + 0] * C2] +
        wgt[r * 3 + 1] * P2[(long)idx[r * 3 + 1] * C2] +
        wgt[r * 3 + 2] * P2[(long)idx[r * 3 + 2] * C2];
  }
  X[t] = v;
}

// ---------------------------------------------------------------- row-wise log_softmax (in place)
__global__ void pn2_log_softmax(float* __restrict__ Y, long M, int O) {
  long t = (long)blockIdx.x * blockDim.x + threadIdx.x;
  if (t >= M) return;
  float* p = Y + t * O;
  float mx = p[0];
  for (int o = 1; o < O; ++o) mx = fmaxf(mx, p[o]);
  float s = 0.0f;
  for (int o = 0; o < O; ++o) s += expf(p[o] - mx);
  float l = logf(s);
  for (int o = 0; o < O; ++o) p[o] = p[o] - mx - l;
}

// ================================================================ host side
struct Layer { const float *w, *b, *g, *beta; int cin, cout; };

static inline int g1(long n, int b) { return (int)((n + b - 1) / b); }

static void launch_gemm(hipStream_t st, const float* X, const float* W, const float* bias,
                        float* Y, long M, int C, int O) {
  int tilesN = (O + 15) >> 4;
  if (tilesN % 4 == 0) {
    long tiles = (M >> 4) * (long)(tilesN / 4);
    pn2_gemm_bias<4><<<g1(tiles, 8), 256, 0, st>>>(X, W, bias, Y, M, C, O);
  } else if (tilesN % 2 == 0) {
    long tiles = (M >> 4) * (long)(tilesN / 2);
    pn2_gemm_bias<2><<<g1(tiles, 8), 256, 0, st>>>(X, W, bias, Y, M, C, O);
  } else {
    long tiles = (M >> 4) * (long)tilesN;
    pn2_gemm_bias<1><<<g1(tiles, 8), 256, 0, st>>>(X, W, bias, Y, M, C, O);
  }
}

static void run_mlp(hipStream_t st, const Layer* Ls, int nL, const float* X0,
                    float* arenaA, float* arenaB, float* mean, float* var,
                    long M, float** Yout) {
  const float* in = X0;
  float* bufs[2] = { arenaA, arenaB };
  int cur = 0;
  for (int i = 0; i < nL; ++i) {
    float* Y = bufs[cur];
    int O = Ls[i].cout, C = Ls[i].cin;   // C % 4 == 0 guaranteed by caller
    launch_gemm(st, in, Ls[i].w, Ls[i].b, Y, M, C, O);
    pn2_bn_stats<<<O, 256, 0, st>>>(Y, mean, var, M, O);
    pn2_bn_relu<<<g1(M * O, 256), 256, 0, st>>>(Y, mean, var, Ls[i].g, Ls[i].beta, M, O);
    in = Y;
    cur ^= 1;
  }
  *Yout = (float*)in;
}

static void run_sa(hipStream_t st, const float* xyz_in, const float* feat_in,
                   int B, int N, int S, int K, float radius, int Cp,
                   const Layer* Ls, int nL,
                   float* new_xyz, float* out_feat,
                   int* fps_i, int* ball_i, float* arenaA, float* arenaB,
                   float* wpad, float* mean, float* var) {
  pn2_fps<<<B, 256, 0, st>>>(xyz_in, fps_i, N, S);
  pn2_gather3<<<g1((long)B * S * 3, 256), 256, 0, st>>>(xyz_in, fps_i, new_xyz, N, S, B);
  pn2_ball<<<g1((long)B * S, 128), 128, 0, st>>>(xyz_in, new_xyz, ball_i, B, N, S, K,
                                                 radius * radius);
  long M = (long)B * S * K;
  int C = 3 + Cp;
  int Cpad = (C + 3) & ~3;
  pn2_group<<<g1(M * Cpad, 256), 256, 0, st>>>(xyz_in, new_xyz, feat_in, ball_i, arenaB,
                                               B, N, S, K, Cp, Cpad);
  Layer L[4];
  for (int i = 0; i < nL; ++i) L[i] = Ls[i];
  if (Cpad != C) {
    pn2_pad_w<<<g1((long)Ls[0].cout * Cpad, 256), 256, 0, st>>>(Ls[0].w, wpad,
                                                                Ls[0].cout, C,
                                                                Ls[0].cout, Cpad);
    L[0].w = wpad;
  }
  L[0].cin = Cpad;
  float* Yfin = nullptr;
  run_mlp(st, L, nL, arenaB, arenaA, arenaB, mean, var, M, &Yfin);
  int Cout = Ls[nL - 1].cout;
  pn2_maxpool<<<g1((long)B * S * Cout, 256), 256, 0, st>>>(Yfin, out_feat, (long)B * S,
                                                           K, Cout);
}

static void run_fp(hipStream_t st, const float* xyz1, const float* xyz2,
                   const float* p1, const float* p2,
                   int B, int N1, int N2, int C1, int C2,
                   const Layer* Ls, int nL, float* out_copy,
                   int* knn_i, float* knn_w, float* arenaA, float* arenaB,
                   float* mean, float* var, float** keep) {
  pn2_knn3<<<g1((long)B * N1, 128), 128, 0, st>>>(xyz1, xyz2, knn_i, knn_w, B, N1, N2);
  int C = C1 + C2;                       // 768/384/320/128: always % 4 == 0
  pn2_interp_concat<<<g1((long)B * N1 * C, 256), 256, 0, st>>>(p1, p2, knn_i, knn_w,
                                                               arenaB, B, N1, N2, C1, C2);
  long M = (long)B * N1;
  float* Yfin = nullptr;
  run_mlp(st, Ls, nL, arenaB, arenaA, arenaB, mean, var, M, &Yfin);
  if (out_copy)
    hipMemcpyAsync(out_copy, Yfin, (size_t)M * Ls[nL - 1].cout * sizeof(float),
                   hipMemcpyDeviceToDevice, st);
  if (keep) *keep = Yfin;
}

extern "C" void kernel_launch(void* const* d_in, const int* in_sizes, int n_in,
                              void* d_out, int out_size, void* d_ws, size_t ws_size,
                              hipStream_t stream) {
  (void)n_in; (void)out_size; (void)ws_size;
  auto F = [&](int i) { return (const float*)d_in[i]; };

  const int sa1c[4] = { 6, 32, 32, 64 };
  const int sa2c[4] = { 67, 64, 64, 128 };
  const int sa3c[4] = { 131, 128, 128, 256 };
  const int sa4c[4] = { 259, 256, 256, 512 };
  const int fp4c[3] = { 768, 256, 256 };
  const int fp3c[3] = { 384, 256, 256 };
  const int fp2c[3] = { 320, 128, 128 };
  const int fp1c[4] = { 128, 128, 128, 128 };

  Layer sa1[3], sa2[3], sa3[3], sa4[3], fp4[2], fp3[2], fp2[2], fp1[3];
  const float *xyz, *c1w, *c1b, *bn1g, *bn1b, *c2w, *c2b;
  int ci = 0;

  if (in_sizes[0] == 16 * 3 * 4096) {
    // setup_inputs() insertion order: xyz, then params; each layer dict {w,b,g,beta}
    xyz = F(ci++);
    auto fill = [&](Layer* L, int n, const int* ch) {
      for (int k = 0; k < n; ++k) {
        L[k].cin = ch[k]; L[k].cout = ch[k + 1];
        L[k].w = F(ci++); L[k].b = F(ci++); L[k].g = F(ci++); L[k].beta = F(ci++);
      }
    };
    fill(sa1, 3, sa1c); fill(sa2, 3, sa2c); fill(sa3, 3, sa3c); fill(sa4, 3, sa4c);
    fill(fp4, 2, fp4c); fill(fp3, 2, fp3c); fill(fp2, 2, fp2c); fill(fp1, 3, fp1c);
    c1w = F(ci++); c1b = F(ci++);
    bn1g = F(ci++); bn1b = F(ci++);
    c2w = F(ci++); c2b = F(ci++);
  } else {
    // jax tree_flatten order: dict keys sorted; layer dicts {b,beta,g,w}; xyz last
    bn1b = F(ci++); bn1g = F(ci++);
    c1b = F(ci++); c1w = F(ci++);
    c2b = F(ci++); c2w = F(ci++);
    auto fill = [&](Layer* L, int n, const int* ch) {
      for (int k = 0; k < n; ++k) {
        L[k].cin = ch[k]; L[k].cout = ch[k + 1];
        L[k].b = F(ci++); L[k].beta = F(ci++); L[k].g = F(ci++); L[k].w = F(ci++);
      }
    };
    fill(fp1, 3, fp1c); fill(fp2, 2, fp2c); fill(fp3, 2, fp3c); fill(fp4, 2, fp4c);
    fill(sa1, 3, sa1c); fill(sa2, 3, sa2c); fill(sa3, 3, sa3c); fill(sa4, 3, sa4c);
    xyz = F(ci++);
  }

  // ---------------- workspace layout (float units)
  float* W = (float*)d_ws;
  size_t off = 0;
  auto alloc = [&](size_t n) { float* p = W + off; off += n; return p; };
  float* pts    = alloc(16 * 4096 * 3);
  float* l1xyz  = alloc(16 * 1024 * 3);
  float* l1f    = alloc((size_t)16 * 1024 * 128);  // 64ch after SA1, 128ch after FP2
  float* l2xyz  = alloc(16 * 256 * 3);
  float* l2f    = alloc((size_t)16 * 256 * 256);   // 128ch after SA2, 256ch after FP3
  float* l3xyz  = alloc(16 * 64 * 3);
  float* l3f    = alloc((size_t)16 * 64 * 256);
  float* l4xyz  = alloc(16 * 16 * 3);
  float* l4f    = alloc((size_t)16 * 16 * 512);
  float* mean   = alloc(512);
  float* var    = alloc(512);
  float* wpad   = alloc(131072);                   // padded weights (max 256*260)
  float* arenaA = alloc((size_t)524288 * 64);      // largest activation (SA1 layer 3)
  float* arenaB = alloc((size_t)524288 * 32);      // X staging + ping-pong
  float* knn_w  = alloc((size_t)16 * 4096 * 3);
  int*   fps_i  = (int*)alloc(16 * 1024);
  int*   ball_i = (int*)alloc((size_t)16 * 1024 * 32);
  int*   knn_i  = (int*)alloc((size_t)16 * 4096 * 3);

  const int B = 16, N0 = 4096;

  pn2_transpose_xyz<<<g1((long)B * N0 * 3, 256), 256, 0, stream>>>(xyz, pts, B, N0);

  // ---------------- set abstraction stack
  run_sa(stream, pts,   pts, B, 4096, 1024, 32, 0.1f, 3,   sa1, 3, l1xyz, l1f,
         fps_i, ball_i, arenaA, arenaB, wpad, mean, var);
  run_sa(stream, l1xyz, l1f, B, 1024, 256,  32, 0.2f, 64,  sa2, 3, l2xyz, l2f,
         fps_i, ball_i, arenaA, arenaB, wpad, mean, var);
  run_sa(stream, l2xyz, l2f, B, 256,  64,   32, 0.4f, 128, sa3, 3, l3xyz, l3f,
         fps_i, ball_i, arenaA, arenaB, wpad, mean, var);
  run_sa(stream, l3xyz, l3f, B, 64,   16,   32, 0.8f, 256, sa4, 3, l4xyz, l4f,
         fps_i, ball_i, arenaA, arenaB, wpad, mean, var);

  // ---------------- feature propagation stack
  run_fp(stream, l3xyz, l4xyz, l3f, l4f, B, 64,   16,  256, 512, fp4, 2, l3f,
         knn_i, knn_w, arenaA, arenaB, mean, var, nullptr);
  run_fp(stream, l2xyz, l3xyz, l2f, l3f, B, 256,  64,  128, 256, fp3, 2, l2f,
         knn_i, knn_w, arenaA, arenaB, mean, var, nullptr);
  run_fp(stream, l1xyz, l2xyz, l1f, l2f, B, 1024, 256, 64,  256, fp2, 2, l1f,
         knn_i, knn_w, arenaA, arenaB, mean, var, nullptr);
  float* l0f = nullptr;  // fp1: p1 = None -> C1 = 0; 3 layers leave result in arenaA
  run_fp(stream, pts, l1xyz, nullptr, l1f, B, 4096, 1024, 0, 128, fp1, 3, nullptr,
         knn_i, knn_w, arenaA, arenaB, mean, var, &l0f);

  // ---------------- head: conv1 + BN + ReLU, conv2 (O=13, W padded to 16 rows), log_softmax
  long M = (long)B * N0;  // 65536
  {
    launch_gemm(stream, l0f, c1w, c1b, arenaB, M, 128, 128);
    pn2_bn_stats<<<128, 256, 0, stream>>>(arenaB, mean, var, M, 128);
    pn2_bn_relu<<<g1(M * 128, 256), 256, 0, stream>>>(arenaB, mean, var, bn1g, bn1b,
                                                      M, 128);
    pn2_pad_w<<<g1(16 * 128, 256), 256, 0, stream>>>(c2w, wpad, 13, 128, 16, 128);
    launch_gemm(stream, arenaB, wpad, c2b, (float*)d_out, M, 128, 13);
    pn2_log_softmax<<<g1(M, 256), 256, 0, stream>>>((float*)d_out, M, 13);
  }
}